// Network_40312563040526
// MI455X (gfx1250) — compile-verified
//
#include <hip/hip_runtime.h>
#include <hip/hip_bf16.h>

typedef float v2f __attribute__((ext_vector_type(2)));
typedef float v8f __attribute__((ext_vector_type(8)));

#define EPSR 1e-8f
#define TSZ 8
#define HW 256

// ---------------------------------------------------------------------------
// Kernel 1: project vertices.  proj[i] = (x_screen, y_screen, z_cam, 0)
// ---------------------------------------------------------------------------
__global__ __launch_bounds__(256)
void project_kernel(const float* __restrict__ verts,
                    const float* __restrict__ Km,
                    const float* __restrict__ Rm,
                    const float* __restrict__ tv,
                    float4* __restrict__ proj, int nV) {
    int i = blockIdx.x * blockDim.x + threadIdx.x;
    if (i >= nV) return;
    float vx = verts[i * 3 + 0], vy = verts[i * 3 + 1], vz = verts[i * 3 + 2];
    float cx = Rm[0] * vx + Rm[1] * vy + Rm[2] * vz + tv[0];
    float cy = Rm[3] * vx + Rm[4] * vy + Rm[5] * vz + tv[1];
    float cz = Rm[6] * vx + Rm[7] * vy + Rm[8] * vz + tv[2];
    float ux = Km[0] * cx + Km[1] * cy + Km[2] * cz;
    float uy = Km[3] * cx + Km[4] * cy + Km[5] * cz;
    proj[i] = make_float4(ux / cz, uy / cz, cz, 0.0f);
}

// ---------------------------------------------------------------------------
// Kernel 2: per-face affine coefficients, written twice:
//  (a) plain float4 per face (for scalar tail + final bary recompute)
//  (b) WMMA A-fragment lane layout per 16-face tile:
//      A*[tile*64 + r*2 + {0,1}]      = (a, b)   for lanes 0..15  (K=0,1)
//      A*[tile*64 + r*2 + 32 + {0,1}] = (c, 0)   for lanes 16..31 (K=2,3)
// Invalid faces get w0 == -1 so the inside test can never pass, and
// inv_z == 0 so they can never beat the EPS-initialized z-scan.
// ---------------------------------------------------------------------------
__global__ __launch_bounds__(256)
void coeff_kernel(const float4* __restrict__ proj,
                  const int* __restrict__ faces,
                  float4* __restrict__ C0,
                  float4* __restrict__ C1,
                  float4* __restrict__ CZ,
                  float* __restrict__ A0,
                  float* __restrict__ A1,
                  float* __restrict__ AZ,
                  int nF, int nF16) {
    int f = blockIdx.x * blockDim.x + threadIdx.x;
    if (f >= nF) return;
    int i0 = faces[f * 3 + 0], i1 = faces[f * 3 + 1], i2 = faces[f * 3 + 2];
    float4 p0 = proj[i0], p1 = proj[i1], p2 = proj[i2];
    float denom = (p1.y - p2.y) * (p0.x - p2.x) + (p2.x - p1.x) * (p0.y - p2.y);
    float zmin  = fminf(p0.z, fminf(p1.z, p2.z));
    bool valid  = (fabsf(denom) > EPSR) && (zmin > EPSR);

    float a0 = 0.0f, b0 = 0.0f, c0 = -1.0f;
    float a1 = 0.0f, b1 = 0.0f, c1 = 0.0f;
    float az = 0.0f, bz = 0.0f, czc = 0.0f;
    if (valid) {
        float inv = 1.0f / denom;
        a0 = (p1.y - p2.y) * inv;  b0 = (p2.x - p1.x) * inv;
        c0 = -a0 * p2.x - b0 * p2.y;
        a1 = (p2.y - p0.y) * inv;  b1 = (p0.x - p2.x) * inv;
        c1 = -a1 * p2.x - b1 * p2.y;
        float iz0 = 1.0f / p0.z, iz1 = 1.0f / p1.z, iz2 = 1.0f / p2.z;
        float r0 = iz0 - iz2, r1 = iz1 - iz2;
        az  = a0 * r0 + a1 * r1;
        bz  = b0 * r0 + b1 * r1;
        czc = c0 * r0 + c1 * r1 + iz2;
    }
    C0[f] = make_float4(a0, b0, c0, 0.0f);
    C1[f] = make_float4(a1, b1, c1, 0.0f);
    CZ[f] = make_float4(az, bz, czc, 0.0f);
    if (f < nF16) {
        int base = (f >> 4) * 64 + (f & 15) * 2;
        A0[base + 0] = a0;  A0[base + 1] = b0;
        A0[base + 32] = c0; A0[base + 33] = 0.0f;
        A1[base + 0] = a1;  A1[base + 1] = b1;
        A1[base + 32] = c1; A1[base + 33] = 0.0f;
        AZ[base + 0] = az;  AZ[base + 1] = bz;
        AZ[base + 32] = czc; AZ[base + 33] = 0.0f;
    }
}

// ---------------------------------------------------------------------------
// Kernel 3: WMMA rasterizer + fused texture sampling.
// One wave32 owns 16 pixels (columns N).  Per 16-face tile:
//   3x b64 A-fragment loads (pre-staged, software pipelined)
//   3x v_wmma_f32_16x16x4_f32  (w0, w1, inv_z)
//   8x {sub,sub,min3,2 cmp,and,2 cndmask} argmax-of-inv_z scan
// Lanes L and L+16 cover face rows 0-7 / 8-15 of the same pixel; one
// shfl_xor(16) merges them.  Barycentrics are recomputed once per pixel.
// ---------------------------------------------------------------------------
__global__ __launch_bounds__(256)
void raster_kernel(const v2f* __restrict__ A0,
                   const v2f* __restrict__ A1,
                   const v2f* __restrict__ AZ,
                   const float4* __restrict__ C0,
                   const float4* __restrict__ C1,
                   const float4* __restrict__ CZ,
                   const float4* __restrict__ proj,
                   const int* __restrict__ faces,
                   const float* __restrict__ texture,
                   const float* __restrict__ bgcolor,
                   float* __restrict__ out, int nF, int nF16) {
    int tid  = blockIdx.x * blockDim.x + threadIdx.x;
    int wave = tid >> 5;
    int lane = threadIdx.x & 31;
    int col  = lane & 15;     // pixel column within tile
    int hf   = lane >> 4;     // half-wave select
    int pix  = wave * 16 + col;
    float px = (float)(pix & (HW - 1)) + 0.5f;
    float py = (float)(pix >> 8) + 0.5f;

    // B fragment (4x16, column = pixel): rows [px, py, 1, 0].
    v2f bfrag;
    bfrag.x = hf ? 1.0f : px;
    bfrag.y = hf ? 0.0f : py;

    v8f zacc = {0.f, 0.f, 0.f, 0.f, 0.f, 0.f, 0.f, 0.f};

    float bestIz = EPSR;      // folds the inv_z > EPS gate into the scan
    int   bestF  = -1;
    int   mOff   = hf << 3;

    if (nF16 > 0) {
        // A-fragment index for tile starting at face fb:  fb*2 + lane
        v2f aw0 = A0[lane], aw1 = A1[lane], azf = AZ[lane];
        for (int fb = 0; fb < nF16; fb += 16) {
            // stage next tile's fragments (clamped reload on last tile)
            int nfb = fb + 16 < nF16 ? fb + 16 : fb;
            int ni  = (nfb << 1) + lane;
            v2f n0 = A0[ni], n1 = A1[ni], nz = AZ[ni];
            if (fb + 32 < nF16) {
                int pi = ((fb + 32) << 1) + lane;
                __builtin_prefetch(&A0[pi], 0, 3);
                __builtin_prefetch(&A1[pi], 0, 3);
                __builtin_prefetch(&AZ[pi], 0, 3);
            }

            v8f d0 = __builtin_amdgcn_wmma_f32_16x16x4_f32(
                false, aw0, false, bfrag, (short)0, zacc, false, false);
            v8f d1 = __builtin_amdgcn_wmma_f32_16x16x4_f32(
                false, aw1, false, bfrag, (short)0, zacc, false, false);
            v8f dz = __builtin_amdgcn_wmma_f32_16x16x4_f32(
                false, azf, false, bfrag, (short)0, zacc, false, false);

            int fbl = fb + mOff;
#pragma unroll
            for (int j = 0; j < 8; ++j) {
                float w0 = d0[j], w1 = d1[j], iz = dz[j];
                float w2 = 1.0f - w0 - w1;
                float m  = fminf(fminf(w0, w1), w2);
                bool upd = (m >= 0.0f) & (iz > bestIz);
                bestIz = upd ? iz : bestIz;
                bestF  = upd ? (fbl + j) : bestF;
            }
            aw0 = n0; aw1 = n1; azf = nz;
        }
    }

    // Merge the two half-waves (same pixel, face rows 0-7 vs 8-15).
    // On exact ties the lower face index wins (matches reference argmin).
    float oIz = __shfl_xor(bestIz, 16, 32);
    int   oF  = __shfl_xor(bestF, 16, 32);
    bool take = (oIz > bestIz) ||
                ((oIz == bestIz) && ((unsigned)oF < (unsigned)bestF));
    if (take) { bestIz = oIz; bestF = oF; }

    if (hf == 0) {
        // scalar tail for nF not a multiple of 16 (empty for nF=13776)
        for (int f = nF16; f < nF; ++f) {
            float4 q0 = C0[f], q1 = C1[f], qz = CZ[f];
            float w0 = fmaf(q0.x, px, fmaf(q0.y, py, q0.z));
            float w1 = fmaf(q1.x, px, fmaf(q1.y, py, q1.z));
            float iz = fmaf(qz.x, px, fmaf(qz.y, py, qz.z));
            float w2 = 1.0f - w0 - w1;
            float m  = fminf(fminf(w0, w1), w2);
            bool upd = (m >= 0.0f) & (iz > bestIz);
            bestIz = upd ? iz : bestIz;
            bestF  = upd ? f : bestF;
        }

        const int P = HW * HW;
        float r, g, b, alpha;
        if (bestF < 0) {
            r = bgcolor[0] * (1.0f / 255.0f);
            g = bgcolor[1] * (1.0f / 255.0f);
            b = bgcolor[2] * (1.0f / 255.0f);
            alpha = 0.0f;
        } else {
            alpha = 1.0f;
            // recompute barycentrics of the winning face (affine in px,py)
            float4 q0 = C0[bestF], q1 = C1[bestF];
            float bW0 = fmaf(q0.x, px, fmaf(q0.y, py, q0.z));
            float bW1 = fmaf(q1.x, px, fmaf(q1.y, py, q1.z));
            int f3 = bestF * 3;
            float z0 = proj[faces[f3 + 0]].z;
            float z1 = proj[faces[f3 + 1]].z;
            float z2 = proj[faces[f3 + 2]].z;
            float w2 = 1.0f - bW0 - bW1;
            float u0 = bW0 / fmaxf(z0, EPSR);
            float u1 = bW1 / fmaxf(z1, EPSR);
            float u2 = w2  / fmaxf(z2, EPSR);
            float s  = fmaxf(u0 + u1 + u2, EPSR);
            u0 /= s; u1 /= s; u2 /= s;
            // trilinear sample of relu(texture), TS=8 grid
            float pos0 = fminf(fmaxf(u0, 0.0f), 1.0f) * (float)(TSZ - 1);
            float pos1 = fminf(fmaxf(u1, 0.0f), 1.0f) * (float)(TSZ - 1);
            float pos2 = fminf(fmaxf(u2, 0.0f), 1.0f) * (float)(TSZ - 1);
            int ix = min(max((int)floorf(pos0), 0), TSZ - 2);
            int iy = min(max((int)floorf(pos1), 0), TSZ - 2);
            int iz = min(max((int)floorf(pos2), 0), TSZ - 2);
            float fx = pos0 - (float)ix;
            float fy = pos1 - (float)iy;
            float fz = pos2 - (float)iz;
            const float* texf =
                texture + (size_t)bestF * (size_t)(TSZ * TSZ * TSZ * 3);
            float ar = 0.0f, ag = 0.0f, ab = 0.0f;
#pragma unroll
            for (int d0i = 0; d0i < 2; ++d0i)
#pragma unroll
                for (int d1i = 0; d1i < 2; ++d1i)
#pragma unroll
                    for (int d2i = 0; d2i < 2; ++d2i) {
                        int idx = ((ix + d0i) * TSZ * TSZ +
                                   (iy + d1i) * TSZ + (iz + d2i)) * 3;
                        float w = (d0i ? fx : 1.0f - fx) *
                                  (d1i ? fy : 1.0f - fy) *
                                  (d2i ? fz : 1.0f - fz);
                        ar += w * fmaxf(texf[idx + 0], 0.0f);
                        ag += w * fmaxf(texf[idx + 1], 0.0f);
                        ab += w * fmaxf(texf[idx + 2], 0.0f);
                    }
            r = ar; g = ag; b = ab;
        }
        out[pix * 3 + 0] = r;
        out[pix * 3 + 1] = g;
        out[pix * 3 + 2] = b;
        out[3 * P + pix] = alpha;
    }
}

// ---------------------------------------------------------------------------
extern "C" void kernel_launch(void* const* d_in, const int* in_sizes, int n_in,
                              void* d_out, int out_size, void* d_ws, size_t ws_size,
                              hipStream_t stream) {
    const float* vertices = (const float*)d_in[0];
    const float* Km       = (const float*)d_in[1];
    const float* Rm       = (const float*)d_in[2];
    const float* tv       = (const float*)d_in[3];
    const float* bg       = (const float*)d_in[4];
    const float* tex      = (const float*)d_in[5];
    const int*   faces    = (const int*)d_in[6];
    int nV   = in_sizes[0] / 3;
    int nF   = in_sizes[6] / 3;
    int nF16 = nF & ~15;                         // full 16-face tiles

    float*  ws   = (float*)d_ws;
    float4* proj = (float4*)ws;                  // nV*4 floats
    float4* C0   = (float4*)(ws + (size_t)nV * 4);
    float4* C1   = C0 + nF;
    float4* CZ   = C1 + nF;
    float*  A0   = (float*)(CZ + nF);            // (nF16/16)*64 floats each
    float*  A1   = A0 + (size_t)(nF16 / 16) * 64;
    float*  AZ   = A1 + (size_t)(nF16 / 16) * 64;

    project_kernel<<<(nV + 255) / 256, 256, 0, stream>>>(
        vertices, Km, Rm, tv, proj, nV);
    coeff_kernel<<<(nF + 255) / 256, 256, 0, stream>>>(
        proj, faces, C0, C1, CZ, A0, A1, AZ, nF, nF16);

    const int P = HW * HW;                       // 65536 pixels
    int waves  = P / 16;                         // 16 pixels per wave32
    int blocks = waves * 32 / 256;               // 512 blocks x 8 waves
    raster_kernel<<<blocks, 256, 0, stream>>>(
        (const v2f*)A0, (const v2f*)A1, (const v2f*)AZ,
        C0, C1, CZ, proj, faces, tex, bg, (float*)d_out, nF, nF16);
}